// PMFLoss_68676527063484
// MI455X (gfx1250) — compile-verified
//
#include <hip/hip_runtime.h>
#include <hip/hip_bf16.h>

#define U_DIM 8192
#define I_DIM 8192
#define K_DIM 256

typedef __attribute__((ext_vector_type(2))) float v2f;
typedef __attribute__((ext_vector_type(8))) float v8f;

__device__ __forceinline__ float wave_sum(float v) {
#pragma unroll
    for (int off = 16; off > 0; off >>= 1) v += __shfl_xor(v, off, 32);
    return v;
}

// ---------------------------------------------------------------------------
// Main fused kernel: predicted = (P^T ⊙ u) @ v^T via V_WMMA_F32_16X16X4_F32,
// fused masked squared-error vs R. Grid: (I/64, U/128), block = 128 (4 waves).
// Each wave: 2 M-strips (16 rows) x 4 N-tiles (16 cols) = 8 f32 accumulators.
// ---------------------------------------------------------------------------
__global__ __launch_bounds__(128) void pmf_main_kernel(
    const float* __restrict__ R,
    const float* __restrict__ uf,
    const float* __restrict__ vf,
    const float* __restrict__ P,
    float* __restrict__ errPart)
{
    const int tid  = threadIdx.x;
    const int wave = tid >> 5;
    const int lane = tid & 31;
    const int l16  = lane & 15;
    const int hk   = (lane >> 4) << 1;   // k sub-offset: 0 for lanes 0-15, 2 for 16-31

    const int n_base = blockIdx.x * 64;               // item (col) tile base
    const int m_base = blockIdx.y * 128 + wave * 32;  // user (row) base for this wave

    const int m0 = m_base + l16;        // strip 0 row held by this lane (A frag)
    const int m1 = m_base + 16 + l16;   // strip 1 row

    v8f acc[2][4] = {};

    const float* ufp0 = uf + (size_t)m0 * K_DIM + hk;
    const float* ufp1 = uf + (size_t)m1 * K_DIM + hk;
    const float* pp   = P + (size_t)hk * U_DIM;       // P[k][*], k advances by 4
    const float* bp0  = vf + (size_t)(n_base +  0 + l16) * K_DIM + hk;
    const float* bp1  = vf + (size_t)(n_base + 16 + l16) * K_DIM + hk;
    const float* bp2  = vf + (size_t)(n_base + 32 + l16) * K_DIM + hk;
    const float* bp3  = vf + (size_t)(n_base + 48 + l16) * K_DIM + hk;

    for (int k0 = 0; k0 < K_DIM; k0 += 4) {
        // A fragments: A[m][k] = P[k][m] * u[m][k]   (P loads coalesced across lanes)
        float p00 = pp[m0];
        float p01 = pp[U_DIM + m0];
        float p10 = pp[m1];
        float p11 = pp[U_DIM + m1];
        v2f u0 = *(const v2f*)ufp0;
        v2f u1 = *(const v2f*)ufp1;
        v2f a0, a1;
        a0.x = p00 * u0.x;  a0.y = p01 * u0.y;
        a1.x = p10 * u1.x;  a1.y = p11 * u1.y;

        // B fragments: B[k][n] = v[n][k]  (consecutive-k float2 per lane)
        v2f b0 = *(const v2f*)bp0;
        v2f b1 = *(const v2f*)bp1;
        v2f b2 = *(const v2f*)bp2;
        v2f b3 = *(const v2f*)bp3;

        acc[0][0] = __builtin_amdgcn_wmma_f32_16x16x4_f32(false, a0, false, b0, (short)0, acc[0][0], false, false);
        acc[1][0] = __builtin_amdgcn_wmma_f32_16x16x4_f32(false, a1, false, b0, (short)0, acc[1][0], false, false);
        acc[0][1] = __builtin_amdgcn_wmma_f32_16x16x4_f32(false, a0, false, b1, (short)0, acc[0][1], false, false);
        acc[1][1] = __builtin_amdgcn_wmma_f32_16x16x4_f32(false, a1, false, b1, (short)0, acc[1][1], false, false);
        acc[0][2] = __builtin_amdgcn_wmma_f32_16x16x4_f32(false, a0, false, b2, (short)0, acc[0][2], false, false);
        acc[1][2] = __builtin_amdgcn_wmma_f32_16x16x4_f32(false, a1, false, b2, (short)0, acc[1][2], false, false);
        acc[0][3] = __builtin_amdgcn_wmma_f32_16x16x4_f32(false, a0, false, b3, (short)0, acc[0][3], false, false);
        acc[1][3] = __builtin_amdgcn_wmma_f32_16x16x4_f32(false, a1, false, b3, (short)0, acc[1][3], false, false);

        ufp0 += 4; ufp1 += 4;
        pp   += (size_t)4 * U_DIM;
        bp0  += 4; bp1 += 4; bp2 += 4; bp3 += 4;
    }

    // Fused masked squared-error epilogue.
    // C/D layout: VGPR j of lane holds (row = j + 8*(lane>>4), col = lane&15) in tile.
    const int row_h = (lane >> 4) << 3;   // 0 or 8
    float s = 0.0f;
#pragma unroll
    for (int strip = 0; strip < 2; ++strip) {
#pragma unroll
        for (int t = 0; t < 4; ++t) {
            const int col = n_base + t * 16 + l16;
#pragma unroll
            for (int j = 0; j < 8; ++j) {
                const int row = m_base + strip * 16 + row_h + j;
                float r = R[(size_t)row * I_DIM + col];
                float d = r - acc[strip][t][j];
                s += (r != 0.0f) ? d * d : 0.0f;
            }
        }
    }

    s = wave_sum(s);
    __shared__ float smem[4];
    if (lane == 0) smem[wave] = s;
    __syncthreads();
    if (tid == 0) {
        float tot = smem[0] + smem[1] + smem[2] + smem[3];
        errPart[blockIdx.y * gridDim.x + blockIdx.x] = tot;   // fixed order -> deterministic
    }
}

// ---------------------------------------------------------------------------
// Regularizer partials: sum(u^2) + sum(v^2) + sum(P^2), 256 blocks x 256 thr.
// ---------------------------------------------------------------------------
__global__ __launch_bounds__(256) void pmf_reg_kernel(
    const float* __restrict__ uf,
    const float* __restrict__ vf,
    const float* __restrict__ P,
    float* __restrict__ regPart)
{
    const size_t n = (size_t)U_DIM * K_DIM;   // each array: 2M elements
    float s = 0.0f;
    for (size_t i = (size_t)blockIdx.x * 256 + threadIdx.x; i < n; i += 256 * 256) {
        float a = uf[i], b = vf[i], c = P[i];
        s += a * a + b * b + c * c;
    }
    s = wave_sum(s);
    __shared__ float smem[8];
    const int wave = threadIdx.x >> 5;
    if ((threadIdx.x & 31) == 0) smem[wave] = s;
    __syncthreads();
    if (threadIdx.x == 0) {
        float tot = 0.0f;
#pragma unroll
        for (int w = 0; w < 8; ++w) tot += smem[w];
        regPart[blockIdx.x] = tot;
    }
}

// ---------------------------------------------------------------------------
// Final combine: out = 0.5*sum(errPart) + 0.05*sum(regPart). Single block.
// ---------------------------------------------------------------------------
__global__ __launch_bounds__(256) void pmf_final_kernel(
    const float* __restrict__ errPart,
    const float* __restrict__ regPart,
    float* __restrict__ out)
{
    float se = 0.0f, sr = 0.0f;
    for (int i = threadIdx.x; i < 8192; i += 256) se += errPart[i];
    if (threadIdx.x < 256) sr = regPart[threadIdx.x];

    se = wave_sum(se);
    sr = wave_sum(sr);
    __shared__ float sme[8], smr[8];
    const int wave = threadIdx.x >> 5;
    if ((threadIdx.x & 31) == 0) { sme[wave] = se; smr[wave] = sr; }
    __syncthreads();
    if (threadIdx.x == 0) {
        float te = 0.0f, tr = 0.0f;
#pragma unroll
        for (int w = 0; w < 8; ++w) { te += sme[w]; tr += smr[w]; }
        out[0] = 0.5f * te + 0.05f * tr;
    }
}

extern "C" void kernel_launch(void* const* d_in, const int* in_sizes, int n_in,
                              void* d_out, int out_size, void* d_ws, size_t ws_size,
                              hipStream_t stream) {
    // setup_inputs order: 0=R, 1=u_features, 2=v_features, 3=alpha, 4=Y, 5=P, 6=Q
    const float* R  = (const float*)d_in[0];
    const float* uf = (const float*)d_in[1];
    const float* vf = (const float*)d_in[2];
    const float* P  = (const float*)d_in[5];

    float* errPart = (float*)d_ws;          // 8192 floats
    float* regPart = errPart + 8192;        // 256 floats

    dim3 grid(I_DIM / 64, U_DIM / 128);     // (128, 64)
    pmf_main_kernel<<<grid, 128, 0, stream>>>(R, uf, vf, P, errPart);
    pmf_reg_kernel<<<256, 256, 0, stream>>>(uf, vf, P, regPart);
    pmf_final_kernel<<<1, 256, 0, stream>>>(errPart, regPart, (float*)d_out);
}